// MPNN_edge_sparse_ogb_61005715472600
// MI455X (gfx1250) — compile-verified
//
#include <hip/hip_runtime.h>

typedef __attribute__((ext_vector_type(2))) float v2f;
typedef __attribute__((ext_vector_type(4))) float v4f;
typedef __attribute__((ext_vector_type(8))) float v8f;
typedef __attribute__((ext_vector_type(4))) unsigned int v4u;
typedef __attribute__((ext_vector_type(4))) int v4i;
typedef __attribute__((ext_vector_type(8))) int v8i;

#define NN    50000
#define NE    800000
#define DIN   96
#define DDEG  32
#define D     128
#define DH    256
#define DUP   128
#define NTILES (NN / 16)   // 3125
#define WAVES_PER_BLOCK 8

// ---------------------------------------------------------------------------
// Kernel 0: h = concat(x, degrees); agg initialized to h (folds the (1+eps)*h
// self term, eps==0, so after edge scatter agg == z). One float4 per thread.
// ---------------------------------------------------------------------------
__global__ __launch_bounds__(256) void k_pack_h(const float* __restrict__ x,
                                                const float* __restrict__ deg,
                                                float* __restrict__ h,
                                                float* __restrict__ agg) {
    int t = blockIdx.x * blockDim.x + threadIdx.x;       // one float4
    int node = t >> 5;                                   // 32 float4 per row
    int c4   = t & 31;
    if (node >= NN) return;
    v4f v;
    if (c4 < 24) v = ((const v4f*)(x   + (long)node * DIN ))[c4];
    else         v = ((const v4f*)(deg + (long)node * DDEG))[c4 - 24];
    ((v4f*)(h   + (long)node * D))[c4] = v;
    ((v4f*)(agg + (long)node * D))[c4] = v;              // self term
}

// ---------------------------------------------------------------------------
// Kernel 1: edge phase. One wave (32 lanes) per edge, float4 per lane.
// msgs = relu(h[src] + ef[e]); atomicAdd into agg[dst] (L2-resident, 25.6MB).
// ef streams 410MB -> non-temporal loads so it does not evict h/agg from the
// 192MB L2 (the whole gather/scatter working set must stay resident there).
// ---------------------------------------------------------------------------
__global__ __launch_bounds__(256) void k_edges(const float* __restrict__ h,
                                               const float* __restrict__ ef,
                                               const int*   __restrict__ ei,
                                               float* __restrict__ agg) {
    int wid  = blockIdx.x * (blockDim.x >> 5) + (threadIdx.x >> 5);
    int lane = threadIdx.x & 31;
    if (wid >= NE) return;
    int src = ei[wid];
    int dst = ei[NE + wid];
    v4f a = ((const v4f*)(h + (long)src * D))[lane];
    v4f b = __builtin_nontemporal_load(((const v4f*)(ef + (long)wid * D)) + lane);
    v4f m;
    m.x = fmaxf(a.x + b.x, 0.f);
    m.y = fmaxf(a.y + b.y, 0.f);
    m.z = fmaxf(a.z + b.z, 0.f);
    m.w = fmaxf(a.w + b.w, 0.f);
    float* p = agg + (long)dst * D + lane * 4;
    atomicAdd(p + 0, m.x);
    atomicAdd(p + 1, m.y);
    atomicAdd(p + 2, m.z);
    atomicAdd(p + 3, m.w);
}

// ---------------------------------------------------------------------------
// TDM: issue one tensor_load_to_lds moving `nelem` f32 from global to LDS.
// D# per CDNA5 ISA: group0 = {count/flags, lds_addr,
// global_addr[31:0], global_addr[56:32] | type=2}; group1 packs data_size=4B,
// tensor_dim0/tile_dim0 = nelem, dims1 = 1, stride0 = nelem. Groups 2/3 unused.
// Toolchain uses the 6-arg (clang-23 / therock-10.0) builtin form.
// ---------------------------------------------------------------------------
__device__ __forceinline__ void tdm_load_f32(unsigned lds_byte_off,
                                             const void* gptr,
                                             unsigned nelem) {
    unsigned long long ga = (unsigned long long)gptr;
    v4u g0;
    g0[0] = 1u;                                            // count=1, user mode
    g0[1] = lds_byte_off;                                  // lds_addr
    g0[2] = (unsigned)(ga & 0xffffffffu);                  // global_addr[31:0]
    g0[3] = (unsigned)((ga >> 32) & 0x01ffffffu)           // global_addr[56:32]
          | (2u << 30);                                    // type = 2 ("image")
    v8i g1;
    g1[0] = (int)(2u << 16);                               // data_size = 4B
    g1[1] = (int)((nelem & 0xffffu) << 16);                // tensor_dim0[15:0]
    g1[2] = (int)(((nelem >> 16) & 0xffffu))               // tensor_dim0[31:16]
          | (1 << 16);                                     // tensor_dim1 = 1
    g1[3] = (int)((nelem & 0xffffu) << 16);                // tile_dim0 = nelem
    g1[4] = 1;                                             // tile_dim1 = 1
    g1[5] = (int)nelem;                                    // tensor_dim0_stride lo
    g1[6] = 0;                                             // stride0 hi, stride1 lo
    g1[7] = 0;
    v4i g2 = {0, 0, 0, 0};
    v4i g3 = {0, 0, 0, 0};
    v8i g4 = {0, 0, 0, 0, 0, 0, 0, 0};
    __builtin_amdgcn_tensor_load_to_lds(g0, g1, g2, g3, g4, 0);
}

// ---------------------------------------------------------------------------
// Kernel 2: out = relu(z@W1 + b1) @ W2 + b2 via V_WMMA_F32_16X16X4_F32.
// W1 (128KB) + W2 (128KB) staged into LDS by the Tensor Data Mover (wave 0),
// plus 8x 16x16 staging tiles (264KB < 320KB/WGP). One wave per 16-node tile:
// layer-1 tile (C layout) -> ReLU -> LDS relayout to A fragments -> layer 2.
// ---------------------------------------------------------------------------
__global__ __launch_bounds__(256) void k_mlp(const float* __restrict__ z,
                                             const float* __restrict__ W1,
                                             const float* __restrict__ b1,
                                             const float* __restrict__ W2,
                                             const float* __restrict__ b2,
                                             float* __restrict__ out) {
    extern __shared__ float lds[];
    float* w1 = lds;                       // [128][256]
    float* w2 = lds + D * DH;              // [256][128]
    float* stage = w2 + DH * DUP;          // 8 waves * 16*16

    if (threadIdx.x < 32) {                // wave 0 drives the TDM
        tdm_load_f32(0u,                     W1, D * DH);
        tdm_load_f32((unsigned)(D * DH * 4), W2, DH * DUP);
        __builtin_amdgcn_s_wait_tensorcnt(0);
    }
    __syncthreads();

    const int wave = threadIdx.x >> 5;
    const int lane = threadIdx.x & 31;
    const int half = lane >> 4;            // 0: lanes 0-15, 1: lanes 16-31
    const int ln   = lane & 15;

    const int tile = blockIdx.x * WAVES_PER_BLOCK + wave;
    if (tile >= NTILES) return;
    const int rowbase = tile * 16;
    float* st = stage + wave * 16 * 16;
    const float* zrow = z + (long)(rowbase + ln) * D;   // this lane's M row

    // layer-2 accumulators: 8 output col tiles of 16, init with b2
    v8f acc2[8];
    for (int o = 0; o < 8; ++o) {
        float bv = b2[o * 16 + ln];
        #pragma unroll
        for (int j = 0; j < 8; ++j) acc2[o][j] = bv;
    }

    for (int t = 0; t < DH / 16; ++t) {            // 16 layer-1 col tiles
        v8f acc1;
        float b1v = b1[t * 16 + ln];
        #pragma unroll
        for (int j = 0; j < 8; ++j) acc1[j] = b1v;

        for (int k = 0; k < D / 4; ++k) {          // 32 K-steps of 4
            const int kb = k * 4 + 2 * half;
            v2f a, b;
            a.x = zrow[kb + 0];
            a.y = zrow[kb + 1];
            b.x = w1[(kb + 0) * DH + t * 16 + ln];
            b.y = w1[(kb + 1) * DH + t * 16 + ln];
            acc1 = __builtin_amdgcn_wmma_f32_16x16x4_f32(
                false, a, false, b, (short)0, acc1, false, false);
        }

        // ReLU + stage C-layout tile to row-major LDS tile [M][N1]
        #pragma unroll
        for (int j = 0; j < 8; ++j)
            st[(half * 8 + j) * 16 + ln] = fmaxf(acc1[j], 0.f);
        // in-wave DS ordering: stores complete before dependent loads

        #pragma unroll
        for (int kk = 0; kk < 4; ++kk) {           // 16 K-vals of this tile
            v2f a;
            a.x = st[ln * 16 + kk * 4 + 2 * half + 0];
            a.y = st[ln * 16 + kk * 4 + 2 * half + 1];
            const int kb2 = t * 16 + kk * 4 + 2 * half;
            for (int o = 0; o < 8; ++o) {
                v2f b;
                b.x = w2[(kb2 + 0) * DUP + o * 16 + ln];
                b.y = w2[(kb2 + 1) * DUP + o * 16 + ln];
                acc2[o] = __builtin_amdgcn_wmma_f32_16x16x4_f32(
                    false, a, false, b, (short)0, acc2[o], false, false);
            }
        }
    }

    // write out (write-once stream: non-temporal): out[rowbase + M][o*16 + N]
    for (int o = 0; o < 8; ++o) {
        #pragma unroll
        for (int j = 0; j < 8; ++j)
            __builtin_nontemporal_store(
                acc2[o][j],
                out + (long)(rowbase + half * 8 + j) * DUP + o * 16 + ln);
    }
}

// ---------------------------------------------------------------------------
extern "C" void kernel_launch(void* const* d_in, const int* in_sizes, int n_in,
                              void* d_out, int out_size, void* d_ws, size_t ws_size,
                              hipStream_t stream) {
    const float* x   = (const float*)d_in[0];
    const float* deg = (const float*)d_in[1];
    const float* ef  = (const float*)d_in[2];
    const float* W1  = (const float*)d_in[3];
    const float* b1  = (const float*)d_in[4];
    const float* W2  = (const float*)d_in[5];
    const float* b2  = (const float*)d_in[6];
    const int*   ei  = (const int*)d_in[7];
    float* out = (float*)d_out;

    float* h   = (float*)d_ws;                       // [NN][128]  25.6 MB
    float* agg = (float*)d_ws + (size_t)NN * D;      // [NN][128]  25.6 MB (becomes z)

    // 1) pack h; agg = h (self term folded in)
    {
        int threads = NN * (D / 4);
        k_pack_h<<<(threads + 255) / 256, 256, 0, stream>>>(x, deg, h, agg);
    }
    // 2) edge gather + relu + scatter-add (agg -> z)
    {
        int blocks = (NE + WAVES_PER_BLOCK - 1) / WAVES_PER_BLOCK;
        k_edges<<<blocks, 256, 0, stream>>>(h, ef, ei, agg);
    }
    // 3) fused MLP with f32 WMMA, TDM-staged LDS-resident weights
    {
        int blocks = (NTILES + WAVES_PER_BLOCK - 1) / WAVES_PER_BLOCK;   // 391
        size_t shmem = (size_t)(D * DH + DH * DUP + WAVES_PER_BLOCK * 16 * 16) * sizeof(float);
        k_mlp<<<blocks, 256, shmem, stream>>>(agg, W1, b1, W2, b2, out);
    }
}